// Self_Attn_1821066134012
// MI455X (gfx1250) — compile-verified
//
#include <hip/hip_runtime.h>
#include <hip/hip_bf16.h>

// ---------------------------------------------------------------------------
// Self-attention over the BATCH axis (B=32, C=256, W=H=64), f32 end-to-end.
// GEMMs: V_WMMA_F32_16X16X4_F32. LDS staging: GLOBAL_LOAD_ASYNC_TO_LDS_B128
// (ASYNCcnt) when the toolchain exposes the builtins, float4 copies otherwise.
// ---------------------------------------------------------------------------

typedef __attribute__((ext_vector_type(2))) float v2f;
typedef __attribute__((ext_vector_type(8))) float v8f;
typedef __attribute__((ext_vector_type(4))) int v4i;

typedef __attribute__((address_space(1))) v4i gas_v4i;  // global
typedef __attribute__((address_space(3))) v4i las_v4i;  // LDS

#define BSZ   32
#define CCH   256
#define CQCH  64
#define NPOS  4096          // W*H
#define DDIM  (CQCH * NPOS) // 262144, energy reduction length
#define YCOLS ((size_t)CCH * NPOS) // 1048576 columns of the batch-mix GEMM

#if __has_builtin(__builtin_amdgcn_global_load_async_to_lds_b128) && \
    __has_builtin(__builtin_amdgcn_s_wait_asynccnt)
#define ASYNC_LDS_OK 1
#endif

// ---- async (or fallback) 16-byte memory->LDS copy -------------------------
__device__ __forceinline__ void stage16(const float* __restrict__ g,
                                        float* __restrict__ l) {
#ifdef ASYNC_LDS_OK
  __builtin_amdgcn_global_load_async_to_lds_b128(
      (gas_v4i*)(v4i*)g, (las_v4i*)(v4i*)l, 0, 0);
#else
  *(float4*)l = *(const float4*)g;
#endif
}

__device__ __forceinline__ void stage_wait() {
#ifdef ASYNC_LDS_OK
  __builtin_amdgcn_s_wait_asynccnt(0);
#endif
}

// ---- WMMA f32 16x16x4 wrapper --------------------------------------------
__device__ __forceinline__ v8f wmma4(v2f a, v2f b, v8f c) {
  // (neg_a, A, neg_b, B, c_mod, C, reuse_a, reuse_b)
  return __builtin_amdgcn_wmma_f32_16x16x4_f32(false, a, false, b, (short)0, c,
                                               false, false);
}

// A-matrix 16x4 f32 fragment from row-major matrix (row stride ld).
// ISA layout: lanes 0-15 -> M=0..15 ; VGPR i holds K = 2*(lane>>4)+i.
__device__ __forceinline__ v2f load_a(const float* __restrict__ A, int ld,
                                      int row0, int k0) {
  int l = threadIdx.x & 31;
  int m = row0 + (l & 15);
  int k = k0 + 2 * (l >> 4);
  v2f a;
  a.x = A[(size_t)m * ld + k];
  a.y = A[(size_t)m * ld + k + 1];
  return a;
}

// B-matrix 4x16 f32 fragment from row-major [K x ld] matrix.
// ISA layout: lanes (mod 16) -> N ; VGPR i holds K = 2*(lane>>4)+i.
__device__ __forceinline__ v2f load_b(const float* __restrict__ B, int ld,
                                      int k0, int col0) {
  int l = threadIdx.x & 31;
  int n = col0 + (l & 15);
  int k = k0 + 2 * (l >> 4);
  v2f b;
  b.x = B[(size_t)k * ld + n];
  b.y = B[(size_t)(k + 1) * ld + n];
  return b;
}

// ===========================================================================
// Kernel 1: Q = Wq*X0 + bq, K = Wk*X1 + bk   (per batch, 64-position tiles)
// grid = (NPOS/64, BSZ), block = 256 (8 waves), dyn LDS = 128 KB
// ===========================================================================
__global__ __launch_bounds__(256) void qk_kernel(
    const float* __restrict__ x0, const float* __restrict__ x1,
    const float* __restrict__ Wq, const float* __restrict__ bq,
    const float* __restrict__ Wk, const float* __restrict__ bk,
    float* __restrict__ Qws, float* __restrict__ Kws) {
  const int PT = 64;
  extern __shared__ float smem[];
  float* xs0 = smem;                // 256 x 64
  float* xs1 = smem + CCH * PT;     // 256 x 64

  int b = blockIdx.y;
  int p0 = blockIdx.x * PT;
  const float* x0b = x0 + (size_t)b * CCH * NPOS + p0;
  const float* x1b = x1 + (size_t)b * CCH * NPOS + p0;

  // async stage: 16 x 16B chunks per 64-float channel row
  for (int i = threadIdx.x; i < (CCH * PT) / 4; i += 256) {
    int c = i >> 4, pc = (i & 15) * 4;
    stage16(x0b + (size_t)c * NPOS + pc, xs0 + c * PT + pc);
    stage16(x1b + (size_t)c * NPOS + pc, xs1 + c * PT + pc);
  }
  stage_wait();
  __syncthreads();

  int w = threadIdx.x >> 5, l = threadIdx.x & 31;

  for (int which = 0; which < 2; ++which) {
    const float* Wm   = which ? Wk : Wq;
    const float* bias = which ? bk : bq;
    const float* xs   = which ? xs1 : xs0;
    float* ows        = which ? Kws : Qws;

    // 16 output tiles (mt 0..3 x nt 0..3); 8 waves take 2 tiles each
    for (int tt = w; tt < 16; tt += 8) {
      int mt = tt & 3, nt = tt >> 2;
      v8f acc = {};
      for (int kk = 0; kk < CCH; kk += 4) {
        v2f a = load_a(Wm, CCH, mt * 16, kk);
        v2f bf = load_b(xs, PT, kk, nt * 16);
        acc = wmma4(a, bf, acc);
      }
      int n = nt * 16 + (l & 15);
      float* orow = ows + (size_t)b * CQCH * NPOS + (size_t)(p0 + n);
#pragma unroll
      for (int r = 0; r < 8; ++r) {
        int m = mt * 16 + r + 8 * (l >> 4); // output channel o
        orow[(size_t)m * NPOS] = acc[r] + bias[m];
      }
    }
  }
}

// ===========================================================================
// Kernel 2: E[b1,b0] += sum_d K[b1,d]*Q[b0,d]  over a 512-wide d-chunk.
// grid = DDIM/512 = 512, block = 256, dyn LDS = 136 KB. E pre-zeroed.
// ===========================================================================
__global__ __launch_bounds__(256) void energy_kernel(
    const float* __restrict__ Qws, const float* __restrict__ Kws,
    float* __restrict__ E) {
  const int DC = 512;
  extern __shared__ float smem[];
  float* Qs = smem;                  // 32 x 512
  float* Ks = smem + 32 * DC;        // 32 x 512
  float* part = smem + 64 * DC;      // [half(2)][tile(4)][vgpr(8)][lane(32)]

  size_t d0 = (size_t)blockIdx.x * DC;
  for (int i = threadIdx.x; i < (32 * DC) / 4; i += 256) {
    int row = i >> 7, dd = (i & 127) * 4;
    stage16(Qws + (size_t)row * DDIM + d0 + dd, Qs + row * DC + dd);
    stage16(Kws + (size_t)row * DDIM + d0 + dd, Ks + row * DC + dd);
  }
  stage_wait();
  __syncthreads();

  int w = threadIdx.x >> 5, l = threadIdx.x & 31;
  int t = w & 3, half = w >> 2;      // 4 output tiles, 2 k-halves
  int mt = t & 1, nt = t >> 1;

  v8f acc = {};
  for (int kk = half * 256; kk < half * 256 + 256; kk += 4) {
    v2f a = load_a(Ks, DC, mt * 16, kk);          // A = K rows (b1)
    int n = nt * 16 + (l & 15);                    // B[k][n] = Q[n][k]
    int k = kk + 2 * (l >> 4);
    v2f bf;
    bf.x = Qs[(size_t)n * DC + k];
    bf.y = Qs[(size_t)n * DC + k + 1];
    acc = wmma4(a, bf, acc);
  }
  float* myp = part + ((size_t)(half * 4 + t) * 8) * 32;
#pragma unroll
  for (int r = 0; r < 8; ++r) myp[r * 32 + l] = acc[r];
  __syncthreads();

  // combine the two k-halves and atomically accumulate into E
  for (int idx = threadIdx.x; idx < 1024; idx += 256) {
    int tt = idx >> 8, r = (idx >> 5) & 7, ll = idx & 31;
    float v = part[((0 * 4 + tt) * 8 + r) * 32 + ll] +
              part[((1 * 4 + tt) * 8 + r) * 32 + ll];
    int mtt = tt & 1, ntt = tt >> 1;
    int b1 = mtt * 16 + r + 8 * (ll >> 4);
    int b0 = ntt * 16 + (ll & 15);
    atomicAdd(&E[b1 * 32 + b0], v);
  }
}

// ===========================================================================
// Kernel 3: row softmax of E (32x32). One wave per row; writes att to ws
// and to the tuple tail of d_out.
// ===========================================================================
__global__ __launch_bounds__(1024) void softmax_kernel(
    const float* __restrict__ E, float* __restrict__ att,
    float* __restrict__ att_out) {
  int b1 = threadIdx.x >> 5, l = threadIdx.x & 31;
  float v = E[b1 * 32 + l];
  float m = v;
#pragma unroll
  for (int o = 16; o; o >>= 1) m = fmaxf(m, __shfl_xor(m, o, 32));
  float e = __expf(v - m);
  float s = e;
#pragma unroll
  for (int o = 16; o; o >>= 1) s += __shfl_xor(s, o, 32);
  float a = e / s;
  att[b1 * 32 + l] = a;
  att_out[b1 * 32 + l] = a;
}

// ===========================================================================
// Kernel 4: Y[32, 1M] = Att[32x32] @ X0[32, 1M]   (batch-mix GEMM)
// grid = 1M/256 = 4096, block = 256, dyn LDS = 36 KB
// ===========================================================================
__global__ __launch_bounds__(256) void ymix_kernel(
    const float* __restrict__ x0, const float* __restrict__ att,
    float* __restrict__ Y) {
  const int CT = 256;
  extern __shared__ float smem[];
  float* Xs = smem;              // 32 x 256
  float* As = smem + 32 * CT;    // 32 x 32

  size_t col0 = (size_t)blockIdx.x * CT;
  for (int i = threadIdx.x; i < (32 * CT) / 4; i += 256) {
    int row = i >> 6, cc = (i & 63) * 4;
    stage16(x0 + (size_t)row * YCOLS + col0 + cc, Xs + row * CT + cc);
  }
  if (threadIdx.x < 256) {  // 1024 att floats = 256 x 16B chunks
    stage16(att + threadIdx.x * 4, As + threadIdx.x * 4);
  }
  stage_wait();
  __syncthreads();

  int w = threadIdx.x >> 5, l = threadIdx.x & 31;
  for (int nt = 2 * w; nt <= 2 * w + 1; ++nt) {
    for (int mt = 0; mt < 2; ++mt) {
      v8f acc = {};
      for (int kk = 0; kk < 32; kk += 4) {
        v2f a = load_a(As, 32, mt * 16, kk);
        v2f bf = load_b(Xs, CT, kk, nt * 16);
        acc = wmma4(a, bf, acc);
      }
      int n = nt * 16 + (l & 15);
#pragma unroll
      for (int r = 0; r < 8; ++r) {
        int b1 = mt * 16 + r + 8 * (l >> 4);
        Y[(size_t)b1 * YCOLS + col0 + n] = acc[r];
      }
    }
  }
}

// ===========================================================================
// Kernel 5: out[b1] = gamma*(Wv @ Y[b1] + bv) + x1[b1]
// grid = (NPOS/64, BSZ), block = 256, dyn LDS = 64 KB
// ===========================================================================
__global__ __launch_bounds__(256) void out_kernel(
    const float* __restrict__ Y, const float* __restrict__ Wv,
    const float* __restrict__ bv, const float* __restrict__ x1,
    const float* __restrict__ gamma, float* __restrict__ out) {
  const int PT = 64;
  extern __shared__ float smem[];
  float* Ys = smem;  // 256 x 64

  int b1 = blockIdx.y;
  int p0 = blockIdx.x * PT;
  const float* Yb = Y + (size_t)b1 * CCH * NPOS + p0;
  for (int i = threadIdx.x; i < (CCH * PT) / 4; i += 256) {
    int c = i >> 4, pc = (i & 15) * 4;
    stage16(Yb + (size_t)c * NPOS + pc, Ys + c * PT + pc);
  }
  stage_wait();
  __syncthreads();

  float g = gamma[0];
  int w = threadIdx.x >> 5, l = threadIdx.x & 31;

  // 64 tiles: mt 0..15 x nt 0..3 ; each of 8 waves takes 8 tiles
  for (int tt = w; tt < 64; tt += 8) {
    int mt = tt & 15, nt = tt >> 4;
    v8f acc = {};
    for (int kk = 0; kk < CCH; kk += 4) {
      v2f a = load_a(Wv, CCH, mt * 16, kk);
      v2f bf = load_b(Ys, PT, kk, nt * 16);
      acc = wmma4(a, bf, acc);
    }
    int p = p0 + nt * 16 + (l & 15);
#pragma unroll
    for (int r = 0; r < 8; ++r) {
      int c = mt * 16 + r + 8 * (l >> 4);
      size_t idx = (size_t)b1 * CCH * NPOS + (size_t)c * NPOS + p;
      out[idx] = g * (acc[r] + bv[c]) + x1[idx];
    }
  }
}

// ===========================================================================
// Host launch
// ===========================================================================
extern "C" void kernel_launch(void* const* d_in, const int* in_sizes, int n_in,
                              void* d_out, int out_size, void* d_ws,
                              size_t ws_size, hipStream_t stream) {
  const float* x0    = (const float*)d_in[0];
  const float* x1    = (const float*)d_in[1];
  const float* Wq    = (const float*)d_in[2];
  const float* bq    = (const float*)d_in[3];
  const float* Wk    = (const float*)d_in[4];
  const float* bk    = (const float*)d_in[5];
  const float* Wv    = (const float*)d_in[6];
  const float* bv    = (const float*)d_in[7];
  const float* gamma = (const float*)d_in[8];

  float* outp    = (float*)d_out;                    // [32,256,64,64]
  float* att_out = outp + (size_t)BSZ * CCH * NPOS;  // tuple tail: att [32,32]

  // Workspace layout (floats). Y (128 MB) reuses the dead Q/K region.
  float* wsf = (float*)d_ws;
  float* Qws = wsf;                               //  32 MB
  float* Kws = wsf + (size_t)BSZ * CQCH * NPOS;   //  32 MB
  float* Yws = wsf;                               // 128 MB (after E done)
  float* E   = wsf + (size_t)BSZ * YCOLS;         //   4 KB @ 128 MB
  float* att = E + 1024;                          //   4 KB

  // 1) Q/K projections
  qk_kernel<<<dim3(NPOS / 64, BSZ), 256, 2 * CCH * 64 * sizeof(float),
              stream>>>(x0, x1, Wq, bq, Wk, bk, Qws, Kws);

  // 2) energy (zero E first; memset-async is graph-capturable)
  (void)hipMemsetAsync(E, 0, 1024 * sizeof(float), stream);
  energy_kernel<<<dim3(DDIM / 512), 256, (64 * 512 + 2048) * sizeof(float),
                  stream>>>(Qws, Kws, E);

  // 3) softmax over batch axis
  softmax_kernel<<<dim3(1), 1024, 0, stream>>>(E, att, att_out);

  // 4) batch-mix: Y = Att @ X0 (overwrites Q/K region)
  ymix_kernel<<<dim3((int)(YCOLS / 256)), 256,
                (32 * 256 + 1024) * sizeof(float), stream>>>(x0, att, Yws);

  // 5) out = gamma*(Wv@Y + bv) + x1
  out_kernel<<<dim3(NPOS / 64, BSZ), 256, CCH * 64 * sizeof(float), stream>>>(
      Yws, Wv, bv, x1, gamma, outp);
}